// GAT_Model_47012712022518
// MI455X (gfx1250) — compile-verified
//
#include <hip/hip_runtime.h>
#include <hip/hip_bf16.h>
#include <stdint.h>

// ---------------------------------------------------------------------------
// GAT (2-layer, PyG GATConv-style) for MI455X / gfx1250, wave32.
// GEMMs use V_WMMA_F32_16X16X4_F32 (full fp32 matrix pipe) since edge
// aggregation (~870MB/layer of gather + atomic traffic) dominates, not GEMM.
// ---------------------------------------------------------------------------

typedef __attribute__((ext_vector_type(2))) float v2f;
typedef __attribute__((ext_vector_type(8))) float v8f;

#define NNODES 50000
#define NEDGES 800000
#define NCOL 256          // HEADS * CH
#define NHEADS 4
#define CH 64

// ---------------- fp32 WMMA GEMM:  H[M,256] = X[M,K] @ W[K,256] -------------
// One block = 8 waves. Each wave owns two 16x16 output tiles (n0, n0+128)
// sharing one A fragment per K-step. M must be multiple of 16 (50000 = 3125*16).
__global__ __launch_bounds__(256) void gemm_wmma_f32(
    const float* __restrict__ X, const float* __restrict__ W,
    float* __restrict__ H, int K) {
  const int wave = threadIdx.x >> 5;
  const int lane = threadIdx.x & 31;
  const int m0 = blockIdx.x * 16;
  const int n0 = wave * 16;        // 0..112
  const int n1 = n0 + 128;         // 128..240

  // A fragment addressing (ISA 7.12.2, 32-bit A 16x4):
  //   lanes 0-15: row = m0+lane, K = k+0,k+1 ; lanes 16-31: K = k+2,k+3
  const int arow = m0 + (lane & 15);
  const int koff = (lane >> 4) * 2;
  const int bcol = lane & 15;

  v8f acc0 = {};
  v8f acc1 = {};
  for (int k = 0; k < K; k += 4) {
    v2f a;
    a.x = X[(size_t)arow * K + (k + koff)];
    a.y = X[(size_t)arow * K + (k + koff + 1)];
    // B fragment (4x16): VGPR r holds row K=k+r (lanes 0-15) / K=k+r+2 (16-31)
    v2f b0, b1;
    b0.x = W[(size_t)(k + koff) * NCOL + n0 + bcol];
    b0.y = W[(size_t)(k + koff + 1) * NCOL + n0 + bcol];
    b1.x = W[(size_t)(k + koff) * NCOL + n1 + bcol];
    b1.y = W[(size_t)(k + koff + 1) * NCOL + n1 + bcol];
    acc0 = __builtin_amdgcn_wmma_f32_16x16x4_f32(
        false, a, false, b0, (short)0, acc0, false, false);
    acc1 = __builtin_amdgcn_wmma_f32_16x16x4_f32(
        false, a, false, b1, (short)0, acc1, false, false);
  }
  // C/D layout: VGPR r -> row m0+r (lanes 0-15) / m0+r+8 (lanes 16-31), col = lane&15
  const int rbase = m0 + ((lane >> 4) << 3);
  const int cix = lane & 15;
#pragma unroll
  for (int r = 0; r < 8; ++r) {
    H[(size_t)(rbase + r) * NCOL + n0 + cix] = acc0[r];
    H[(size_t)(rbase + r) * NCOL + n1 + cix] = acc1[r];
  }
}

// ---------------- attention projections: alpha_s/d[n,h] = <h[n,h,:], a> ----
__global__ void alpha_kernel(const float* __restrict__ H,
                             const float* __restrict__ a_src,
                             const float* __restrict__ a_dst,
                             float* __restrict__ as_, float* __restrict__ ad_) {
  int idx = blockIdx.x * blockDim.x + threadIdx.x;   // node*4 + head
  if (idx >= NNODES * NHEADS) return;
  int head = idx & 3;
  int node = idx >> 2;
  const float* hp = H + (size_t)node * NCOL + head * CH;
  const float* s = a_src + head * CH;
  const float* d = a_dst + head * CH;
  float ss = 0.f, dd = 0.f;
#pragma unroll 8
  for (int c = 0; c < CH; ++c) {
    float v = hp[c];
    ss += v * s[c];
    dd += v * d[c];
  }
  as_[idx] = ss;
  ad_[idx] = dd;
}

// ---------------- zero accumulator + reset max/denom -----------------------
__global__ void init_kernel(float* __restrict__ out, unsigned* __restrict__ mx,
                            float* __restrict__ den) {
  int i = blockIdx.x * blockDim.x + threadIdx.x;
  if (i < NNODES * NCOL) out[i] = 0.f;
  if (i < NNODES * NHEADS) { mx[i] = 0u; den[i] = 0.f; }
}

// ordered-uint mapping: monotonic float -> u32 (so integer atomicMax == fmax)
__device__ __forceinline__ unsigned f2ord(float f) {
  unsigned u = __float_as_uint(f);
  return (u >> 31) ? ~u : (u | 0x80000000u);
}
__device__ __forceinline__ float ord2f(unsigned u) {
  return (u >> 31) ? __uint_as_float(u & 0x7fffffffu) : __uint_as_float(~u);
}

__device__ __forceinline__ void edge_sd(int e, const long long* __restrict__ srcp,
                                        const long long* __restrict__ dstp,
                                        int& s, int& d) {
  if (e < NEDGES) { s = (int)srcp[e]; d = (int)dstp[e]; }
  else            { s = d = e - NEDGES; }   // appended self-loops
}

// ---------------- edge pass 1: segment max of leaky_relu(e) ---------------
__global__ void edge_max_kernel(const long long* __restrict__ srcp,
                                const long long* __restrict__ dstp,
                                const float* __restrict__ as_,
                                const float* __restrict__ ad_,
                                unsigned* __restrict__ mx) {
  int idx = blockIdx.x * blockDim.x + threadIdx.x;   // edge*4 + head
  if (idx >= (NEDGES + NNODES) * NHEADS) return;
  int head = idx & 3;
  int e = idx >> 2;
  int s, d;
  edge_sd(e, srcp, dstp, s, d);
  float v = as_[s * 4 + head] + ad_[d * 4 + head];
  v = v > 0.f ? v : 0.2f * v;                         // leaky_relu(0.2)
  atomicMax(&mx[d * 4 + head], f2ord(v));
}

// ---------------- edge pass 2: denom = segment sum of exp(e - m) -----------
__global__ void edge_denom_kernel(const long long* __restrict__ srcp,
                                  const long long* __restrict__ dstp,
                                  const float* __restrict__ as_,
                                  const float* __restrict__ ad_,
                                  const unsigned* __restrict__ mx,
                                  float* __restrict__ den) {
  int idx = blockIdx.x * blockDim.x + threadIdx.x;
  if (idx >= (NEDGES + NNODES) * NHEADS) return;
  int head = idx & 3;
  int e = idx >> 2;
  int s, d;
  edge_sd(e, srcp, dstp, s, d);
  float v = as_[s * 4 + head] + ad_[d * 4 + head];
  v = v > 0.f ? v : 0.2f * v;
  atomicAdd(&den[d * 4 + head], __expf(v - ord2f(mx[d * 4 + head])));
}

// ---------------- edge pass 3: out[dst] += alpha * h[src] ------------------
// One wave32 per edge; 8 channels/lane (c = lane + 32k -> head = k>>1, constant
// after unroll). fp32 global atomic adds.
__global__ __launch_bounds__(256) void edge_agg_kernel(
    const long long* __restrict__ srcp, const long long* __restrict__ dstp,
    const float* __restrict__ as_, const float* __restrict__ ad_,
    const unsigned* __restrict__ mx, const float* __restrict__ den,
    const float* __restrict__ H, float* __restrict__ out) {
  int wid = (int)((blockIdx.x * blockDim.x + threadIdx.x) >> 5);
  int lane = threadIdx.x & 31;
  if (wid >= NEDGES + NNODES) return;
  int s, d;
  edge_sd(wid, srcp, dstp, s, d);
  float alpha[NHEADS];
#pragma unroll
  for (int hd = 0; hd < NHEADS; ++hd) {
    float v = as_[s * 4 + hd] + ad_[d * 4 + hd];
    v = v > 0.f ? v : 0.2f * v;
    float ex = __expf(v - ord2f(mx[d * 4 + hd]));
    alpha[hd] = ex / (den[d * 4 + hd] + 1e-16f);
  }
  const float* hs = H + (size_t)s * NCOL;
  float* od = out + (size_t)d * NCOL;
#pragma unroll
  for (int k = 0; k < 8; ++k) {
    int c = lane + k * 32;
    atomicAdd(&od[c], alpha[k >> 1] * hs[c]);
  }
}

// ---------------- bias + ELU (in-place on accumulator) ---------------------
__global__ void bias_elu_kernel(float* __restrict__ out,
                                const float* __restrict__ b) {
  int i = blockIdx.x * blockDim.x + threadIdx.x;
  if (i >= NNODES * NCOL) return;
  float v = out[i] + b[i & (NCOL - 1)];
  out[i] = v > 0.f ? v : (__expf(v) - 1.f);
}

// ---------------- final FC (256->1) + sigmoid, wave per node ---------------
__global__ __launch_bounds__(256) void fc_kernel(const float* __restrict__ act,
                                                 const float* __restrict__ w,
                                                 const float* __restrict__ b,
                                                 float* __restrict__ y) {
  int wid = (int)((blockIdx.x * blockDim.x + threadIdx.x) >> 5);
  int lane = threadIdx.x & 31;
  if (wid >= NNODES) return;
  const float* a = act + (size_t)wid * NCOL;
  float sum = 0.f;
#pragma unroll
  for (int k = 0; k < 8; ++k) {
    int c = lane + k * 32;
    sum += a[c] * w[c];
  }
#pragma unroll
  for (int off = 16; off > 0; off >>= 1) sum += __shfl_down(sum, off, 32);
  if (lane == 0) {
    float v = sum + b[0];
    y[wid] = 1.f / (1.f + __expf(-v));
  }
}

// ---------------------------------------------------------------------------
extern "C" void kernel_launch(void* const* d_in, const int* in_sizes, int n_in,
                              void* d_out, int out_size, void* d_ws, size_t ws_size,
                              hipStream_t stream) {
  (void)in_sizes; (void)n_in; (void)out_size; (void)ws_size;
  const float*     x      = (const float*)d_in[0];
  const long long* ei     = (const long long*)d_in[1];   // int64 [2, E]
  const float*     W1     = (const float*)d_in[2];
  const float*     a_src1 = (const float*)d_in[3];
  const float*     a_dst1 = (const float*)d_in[4];
  const float*     b1     = (const float*)d_in[5];
  const float*     W2     = (const float*)d_in[6];
  const float*     a_src2 = (const float*)d_in[7];
  const float*     a_dst2 = (const float*)d_in[8];
  const float*     b2     = (const float*)d_in[9];
  const float*     fc_w   = (const float*)d_in[10];
  const float*     fc_b   = (const float*)d_in[11];
  float* y = (float*)d_out;

  const long long* srcp = ei;
  const long long* dstp = ei + NEDGES;

  // workspace layout (~106 MB)
  float*    h   = (float*)d_ws;                          // [N,256]
  float*    out = h   + (size_t)NNODES * NCOL;           // [N,256]
  float*    as_ = out + (size_t)NNODES * NCOL;           // [N,4]
  float*    ad_ = as_ + (size_t)NNODES * NHEADS;         // [N,4]
  unsigned* mx  = (unsigned*)(ad_ + (size_t)NNODES * NHEADS);   // [N,4]
  float*    den = (float*)(mx + (size_t)NNODES * NHEADS);       // [N,4]

  const int ETOT = NEDGES + NNODES;
  const int gInit  = (NNODES * NCOL + 255) / 256;
  const int gNH    = (NNODES * NHEADS + 255) / 256;
  const int gEdgeH = (ETOT * NHEADS + 255) / 256;
  const int gEdgeW = (ETOT + 7) / 8;          // 8 waves/block, 1 wave/edge
  const int gNodeW = (NNODES + 7) / 8;

  // ---------------- layer 1 ----------------
  gemm_wmma_f32<<<NNODES / 16, 256, 0, stream>>>(x, W1, h, 128);
  init_kernel<<<gInit, 256, 0, stream>>>(out, mx, den);
  alpha_kernel<<<gNH, 256, 0, stream>>>(h, a_src1, a_dst1, as_, ad_);
  edge_max_kernel<<<gEdgeH, 256, 0, stream>>>(srcp, dstp, as_, ad_, mx);
  edge_denom_kernel<<<gEdgeH, 256, 0, stream>>>(srcp, dstp, as_, ad_, mx, den);
  edge_agg_kernel<<<gEdgeW, 256, 0, stream>>>(srcp, dstp, as_, ad_, mx, den, h, out);
  bias_elu_kernel<<<gInit, 256, 0, stream>>>(out, b1);   // out = act2

  // ---------------- layer 2 ----------------
  gemm_wmma_f32<<<NNODES / 16, 256, 0, stream>>>(out, W2, h, 256);
  init_kernel<<<gInit, 256, 0, stream>>>(out, mx, den);  // act2 consumed; reuse
  alpha_kernel<<<gNH, 256, 0, stream>>>(h, a_src2, a_dst2, as_, ad_);
  edge_max_kernel<<<gEdgeH, 256, 0, stream>>>(srcp, dstp, as_, ad_, mx);
  edge_denom_kernel<<<gEdgeH, 256, 0, stream>>>(srcp, dstp, as_, ad_, mx, den);
  edge_agg_kernel<<<gEdgeW, 256, 0, stream>>>(srcp, dstp, as_, ad_, mx, den, h, out);
  bias_elu_kernel<<<gInit, 256, 0, stream>>>(out, b2);

  // ---------------- head ----------------
  fc_kernel<<<gNodeW, 256, 0, stream>>>(out, fc_w, fc_b, y);
}